// NSGT_4071628996940
// MI455X (gfx1250) — compile-verified
//
#include <hip/hip_runtime.h>
#include <math.h>

// ---------------------------------------------------------------------------
// NSGT matrixform on MI455X (gfx1250, wave32).
//
// Sizes (fixed by the reference _design()):
//   LS    = 1323000 = 2^3 * 3^3 * 5^3 * 7^2   (forward FFT length, batch 2)
//   nbins = 98, Mmax = 123000 = 41 * 25 * 15 * 8  (inverse FFT length, batch 196)
//
// Memory-bound workload (~2.1 GB streamed vs ~2.5 GFLOP) => use full-precision
// V_WMMA_F32_16X16X4_F32 for the only dense-GEMM-shaped piece: the twiddle-free
// first Stockham stage of the IFFT (DFT_41 x 588000 columns). The 36 KB DFT
// matrix is staged into LDS by the Tensor Data Mover (tensor_load_to_lds).
// ---------------------------------------------------------------------------

#define PI_F 3.14159265358979323846f

typedef float v2f __attribute__((ext_vector_type(2)));
typedef float v8f __attribute__((ext_vector_type(8)));
typedef unsigned int u32x4 __attribute__((ext_vector_type(4)));
typedef unsigned int u32x8 __attribute__((ext_vector_type(8)));

#define LSN    1323000
#define NBINS  98
#define MLEN   123000
#define NBATCH (2 * NBINS)      // 196 rows to inverse-transform
#define NCOL   (MLEN / 41)      // 3000 columns per row for the radix-41 GEMM
#define NGRP   188              // ceil(3000 / 16) column groups of 16
#define MATN   (96 * 96)        // padded real-expanded DFT_41 matrix elements

// ---------------------------------------------------------------------------
// pack: real signal -> interleaved complex
// ---------------------------------------------------------------------------
__global__ __launch_bounds__(256) void k_pack(const float* __restrict__ s,
                                              float2* __restrict__ o,
                                              long long n) {
    long long i = (long long)blockIdx.x * blockDim.x + threadIdx.x;
    if (i >= n) return;
    o[i] = make_float2(s[i], 0.0f);
}

// ---------------------------------------------------------------------------
// Generic Stockham autosort stage, radix R (Govindaraju formulation).
//   Ns = product of radices already applied (1 on first stage).
//   sign = -1 forward, +1 inverse.  scale applied at store (1/N on last stage).
// Fully unrolled so v[], w[], root[] stay in VGPRs (constant indexing).
// ---------------------------------------------------------------------------
template <int R>
__global__ __launch_bounds__(128) void k_stockham(const float2* __restrict__ in,
                                                  float2* __restrict__ out,
                                                  int N, int Ns, int nbatch,
                                                  float sign, float scale) {
    const int nR = N / R;
    long long tid   = (long long)blockIdx.x * blockDim.x + threadIdx.x;
    long long total = (long long)nR * nbatch;
    if (tid >= total) return;
    int j = (int)(tid % nR);
    int b = (int)(tid / nR);
    const float2* x = in  + (long long)b * N;
    float2*       y = out + (long long)b * N;

    // R-th roots of unity (signed) kept in registers.
    float2 root[R];
#pragma unroll
    for (int k = 0; k < R; ++k) {
        float a = sign * (2.0f * PI_F) * (float)k / (float)R;
        __sincosf(a, &root[k].y, &root[k].x);   // x=cos, y=sin
    }

    const int jNs = j % Ns;
    const int NsR = Ns * R;

    // Load + stage twiddle: v[r] = x[j + r*N/R] * exp(sign*2pi*i*(jNs*r)/NsR)
    float2 v[R];
#pragma unroll
    for (int r = 0; r < R; ++r) {
        float2 t = x[j + (long long)r * nR];
        int ph = (int)(((long long)jNs * r) % NsR);     // keep sincos arg small
        float a = sign * (2.0f * PI_F) * (float)ph / (float)NsR;
        float sn, cs;
        __sincosf(a, &sn, &cs);
        v[r] = make_float2(t.x * cs - t.y * sn, t.x * sn + t.y * cs);
    }

    // In-register R-point DFT: w[t] = sum_r v[r] * root[(t*r) % R]
    float2 w[R];
#pragma unroll
    for (int t = 0; t < R; ++t) {
        float2 acc = v[0];
#pragma unroll
        for (int r = 1; r < R; ++r) {
            float2 rt = root[(t * r) % R];
            acc.x += v[r].x * rt.x - v[r].y * rt.y;
            acc.y += v[r].x * rt.y + v[r].y * rt.x;
        }
        w[t] = acc;
    }

    long long idxD = (long long)(j / Ns) * NsR + jNs;
#pragma unroll
    for (int t = 0; t < R; ++t)
        y[idxD + (long long)t * Ns] = make_float2(w[t].x * scale, w[t].y * scale);
}

// ---------------------------------------------------------------------------
// Gather + window: t[b,j,m] = ft[b, idx[j,m]] * gw[j,m]  (writes d_out buffer)
// ---------------------------------------------------------------------------
__global__ __launch_bounds__(256) void k_gather(const float2* __restrict__ ft,
                                                const int* __restrict__ idx,
                                                const float* __restrict__ gw,
                                                float2* __restrict__ t) {
    long long i = (long long)blockIdx.x * blockDim.x + threadIdx.x;
    const long long per   = (long long)NBINS * MLEN;
    const long long total = 2LL * per;
    if (i >= total) return;
    long long jm = i % per;
    int b = (int)(i / per);
    int id  = idx[jm];
    float g = gw[jm];
    float2 f = ft[(long long)b * LSN + id];
    t[i] = make_float2(f.x * g, f.y * g);
}

// ---------------------------------------------------------------------------
// Build the real-expanded DFT_41 matrix, zero-padded to 96x96 (row-major).
// Complex D[t][r] = exp(+2*pi*i*t*r/41)  (inverse sign) expands as
//   M[2t  ][2r] =  c   M[2t  ][2r+1] = -s
//   M[2t+1][2r] =  s   M[2t+1][2r+1] =  c
// so M * (interleaved re/im vector) performs the complex matvec natively.
// ---------------------------------------------------------------------------
__global__ __launch_bounds__(256) void k_buildmat(float* __restrict__ M) {
    int i = blockIdx.x * blockDim.x + threadIdx.x;
    if (i >= MATN) return;
    int row = i / 96, col = i % 96;
    int t = row >> 1, r = col >> 1;
    float val = 0.0f;
    if (t < 41 && r < 41) {
        float a = (2.0f * PI_F) * (float)((t * r) % 41) / 41.0f;  // +sign (ifft)
        float sn, cs;
        __sincosf(a, &sn, &cs);
        int pr = row & 1, pc = col & 1;
        val = (pr == pc) ? cs : (pr ? sn : -sn);
    }
    M[i] = val;
}

// ---------------------------------------------------------------------------
// Radix-41 first IFFT stage as a WMMA f32 GEMM.
// Stockham stage with Ns=1 has unity twiddles, so per (batch row b, column j):
//   Y[t] = sum_r D41[t][r] * X[r],  X[r] = in[b][j + 3000*r], out[b][41*j + t].
// Real-expanded: out96 = M96x96 * in96 over 16-column groups per wave.
// A = M tile (16x4, LDS, staged by TDM), B = input tile (4x16, one float2
// load / lane), C/D = 6 accumulators of v8f (96 output rows); 24 K-steps.
// ---------------------------------------------------------------------------
__global__ __launch_bounds__(256) void k_dft41_wmma(const float2* __restrict__ xin,
                                                    float2* __restrict__ yout,
                                                    const float* __restrict__ Mmat) {
    __shared__ float Msh[MATN];

    // Stage the 36 KB DFT matrix into LDS with the Tensor Data Mover:
    // one scalar-issued DMA instead of 36 vector loads per thread.
    // D# group0: count=1, lds_addr=&Msh, global_addr=Mmat, type=2 ("image").
    // D# group1: data_size=4B (code 2), 2-D tensor 9216x1, tile 9216x1,
    //            dim0 stride 9216; no multicast / padding / iteration.
    // NOTE: &Msh is passed as an asm operand so the LDS object escapes and the
    // "memory" clobber makes the DMA's LDS write visible to the optimizer
    // (otherwise the Msh loads below get folded to undef and deleted).
    float* mshp = &Msh[0];
    if ((threadIdx.x >> 5) == 0) {          // wave 0 issues; TDM ignores EXEC
        unsigned long long ga = (unsigned long long)(const void*)Mmat;
        // Low 32 bits of a flat pointer into the LDS aperture are the LDS
        // byte offset.
        unsigned int lds_off = (unsigned int)(unsigned long long)(void*)mshp;
        u32x4 g0;
        g0[0] = 1u;                                   // count=1, user mode
        g0[1] = lds_off;                              // lds_addr (bytes)
        g0[2] = (unsigned)(ga & 0xffffffffu);         // global_addr[31:0]
        g0[3] = (unsigned)((ga >> 32) & 0x01ffffffu)  // global_addr[56:32]
                | (2u << 30);                         // type = 2
        u32x8 g1;
        g1[0] = 0x00020000u;                  // wg_mask=0, data_size=2 (4B)
        g1[1] = (MATN & 0xffffu) << 16;       // tensor_dim0[15:0]
        g1[2] = (MATN >> 16) | (1u << 16);    // tensor_dim0[31:16], tensor_dim1=1
        g1[3] = ((unsigned)MATN) << 16;       // tile_dim0 = 9216
        g1[4] = 1u;                           // tile_dim1 = 1, tile_dim2 = 0
        g1[5] = (unsigned)MATN;               // tensor_dim0_stride[31:0]
        g1[6] = 0u;                           // stride hi / dim1_stride lo
        g1[7] = 0u;
        asm volatile("tensor_load_to_lds %0, %1"
                     :: "s"(g0), "s"(g1), "r"(mshp) : "memory");
        __builtin_amdgcn_s_wait_tensorcnt(0); // DMA complete before publishing
    }
    __syncthreads();

    const int wid  = blockIdx.x * (blockDim.x >> 5) + (threadIdx.x >> 5);
    const int lane = threadIdx.x & 31;
    const int total = NBATCH * NGRP;
    if (wid >= total) return;               // wave-uniform: EXEC all-ones inside

    const int b  = wid / NGRP;
    const int j0 = (wid % NGRP) * 16;

    const float2* xb = xin  + (long long)b * MLEN;
    float2*       yb = yout + (long long)b * MLEN;

    const int hi = lane >> 4;               // 0 | 1 (half-wave)
    const int lo = lane & 15;
    const int col = j0 + lo;
    const bool colok = (col < NCOL);

    v8f acc[6] = {};

    for (int kk = 0; kk < 96; kk += 4) {
        // B tile 4x16: lane holds column `col`, K rows kk+2*hi, kk+2*hi+1
        // => one complex input element (re,im) at r = kk/2 + hi.
        int r = (kk >> 1) + hi;             // 0..47; valid < 41
        v2f bt;
        if (r < 41 && colok) {
            float2 xv = xb[col + NCOL * r];
            bt[0] = xv.x; bt[1] = xv.y;
        } else {
            bt[0] = 0.0f; bt[1] = 0.0f;
        }
        const int ak = kk + 2 * hi;
#pragma unroll
        for (int rt = 0; rt < 6; ++rt) {
            // A tile 16x4: lane row = lo, K cols ak, ak+1 (from LDS)
            v2f at;
            at[0] = mshp[(rt * 16 + lo) * 96 + ak];
            at[1] = mshp[(rt * 16 + lo) * 96 + ak + 1];
            acc[rt] = __builtin_amdgcn_wmma_f32_16x16x4_f32(
                false, at, false, bt, (short)0, acc[rt], false, false);
        }
    }

    // Store: acc[rt][v] is output row i = rt*16 + v + 8*hi, column `col`.
    // Rows (2t, 2t+1) = (re, im) pair -> single float2 store at out[41*col + t].
#pragma unroll
    for (int rt = 0; rt < 6; ++rt) {
#pragma unroll
        for (int v = 0; v < 8; v += 2) {
            int t = rt * 8 + 4 * hi + (v >> 1);
            if (t < 41 && colok)
                yb[(long long)col * 41 + t] = make_float2(acc[rt][v], acc[rt][v + 1]);
        }
    }
}

// ---------------------------------------------------------------------------
// Host orchestration. Workspace layout (bytes):
//   F0   @ 0          : 2*1323000 complex  (21,168,000)
//   F1   @ 21168000   : 2*1323000 complex  (21,168,000)
//   Mmat @ 42336000   : 96*96 f32          (36,864)
//   T1   @ 42373120   : 196*123000 complex (192,864,000)
// d_out doubles as the other IFFT ping-pong buffer (T0); final radix-8 stage
// writes d_out with the 1/123000 ifft scale. Total ws need ~= 235.3 MB.
// ---------------------------------------------------------------------------
extern "C" void kernel_launch(void* const* d_in, const int* in_sizes, int n_in,
                              void* d_out, int out_size, void* d_ws, size_t ws_size,
                              hipStream_t stream) {
    (void)in_sizes; (void)n_in; (void)out_size; (void)ws_size;
    const float* s   = (const float*)d_in[0];
    const int*   idx = (const int*)d_in[1];
    const float* gw  = (const float*)d_in[2];

    char* ws = (char*)d_ws;
    float2* F0 = (float2*)(ws);
    float2* F1 = (float2*)(ws + 21168000);
    float*  Mm = (float*)(ws + 42336000);
    float2* T1 = (float2*)(ws + 42373120);
    float2* T0 = (float2*)d_out;

    // 1) pack real -> complex
    {
        long long n = 2LL * LSN;
        k_pack<<<(unsigned)((n + 255) / 256), 256, 0, stream>>>(s, F0, n);
    }
    // 2) DFT_41 matrix (tiny, once per launch)
    k_buildmat<<<(MATN + 255) / 256, 256, 0, stream>>>(Mm);

    // 3) forward FFT, N = 1323000 = 8*27*25*5*7*7, batch 2, sign -1
    {
        int Ns = 1;
        float2 *pa = F0, *pb = F1;
#define FWD_STAGE(R)                                                          \
        {                                                                     \
            long long tot = (long long)(LSN / (R)) * 2;                       \
            k_stockham<R><<<(unsigned)((tot + 127) / 128), 128, 0, stream>>>( \
                pa, pb, LSN, Ns, 2, -1.0f, 1.0f);                             \
            Ns *= (R);                                                        \
            float2* tmp = pa; pa = pb; pb = tmp;                              \
        }
        FWD_STAGE(8)
        FWD_STAGE(27)
        FWD_STAGE(25)
        FWD_STAGE(5)
        FWD_STAGE(7)
        FWD_STAGE(7)
#undef FWD_STAGE
        // 6 stages (even) -> result back in F0 (== pa)
    }

    // 4) gather * window into T0 (= d_out)
    {
        long long n = 2LL * NBINS * MLEN;
        k_gather<<<(unsigned)((n + 255) / 256), 256, 0, stream>>>(F0, idx, gw, T0);
    }

    // 5) inverse FFT over last axis, M = 123000 = 41*25*15*8, batch 196, sign +1
    //    stage 1: radix-41 via WMMA (Ns=1 -> twiddle-free dense DFT GEMM)
    {
        int waves  = NBATCH * NGRP;
        int blocks = (waves + 7) / 8;           // 8 waves per 256-thread block
        k_dft41_wmma<<<blocks, 256, 0, stream>>>(T0, T1, Mm);
    }
    //    stages 2-4: scalar Stockham; final stage applies 1/M and lands in d_out
    {
        long long tot = (MLEN / 25LL) * NBATCH;
        k_stockham<25><<<(unsigned)((tot + 127) / 128), 128, 0, stream>>>(
            T1, T0, MLEN, 41, NBATCH, 1.0f, 1.0f);
    }
    {
        long long tot = (MLEN / 15LL) * NBATCH;
        k_stockham<15><<<(unsigned)((tot + 127) / 128), 128, 0, stream>>>(
            T0, T1, MLEN, 41 * 25, NBATCH, 1.0f, 1.0f);
    }
    {
        long long tot = (MLEN / 8LL) * NBATCH;
        k_stockham<8><<<(unsigned)((tot + 127) / 128), 128, 0, stream>>>(
            T1, T0, MLEN, 41 * 25 * 15, NBATCH, 1.0f, 1.0f / (float)MLEN);
    }
}